// ExtractPatchesPositionLayer_39015482917580
// MI455X (gfx1250) — compile-verified
//
#include <hip/hip_runtime.h>
#include <stdint.h>

// ExtractPatchesPositionLayer for MI455X (gfx1250).
// Memory-bound bilinear patch extraction: TDM (tensor_load_to_lds) stages each
// 128x128 f32 image into LDS exactly once; a 4-tap stencil with per-(b,c)
// uniform weights (frac of the shift) produces the (B,64,64,4) output with
// coalesced float4 stores. ~268 MB total traffic -> ~11.5us floor @ 23.3 TB/s.

typedef uint32_t u32x4 __attribute__((ext_vector_type(4)));
typedef uint32_t u32x8 __attribute__((ext_vector_type(8)));

#define M_DIM 128
#define N_DIM 64
#define C_DIM 4
#define C0    32          // (M - N) / 2
#define ROWS_PER_WAVE 16  // 8 waves x 16 rows = 128 rows

__global__ __launch_bounds__(256)
void extract_patches_tdm_kernel(const float* __restrict__ img,   // (B,128,128,1)
                                const float* __restrict__ pos,   // (B,1,2,4)
                                float* __restrict__ out)         // (B,64,64,4)
{
    __shared__ float tile[M_DIM * M_DIM];   // 64 KB

    const int b    = blockIdx.x;
    const int tid  = threadIdx.x;
    const int wave = tid >> 5;              // 0..7, uniform within a wave

    // ---------------- TDM stage: each wave DMAs a 128x16 f32 slice ----------
    {
        const uint64_t gaddr =
            (uint64_t)(uintptr_t)(img + (size_t)b * (M_DIM * M_DIM)
                                      + (size_t)wave * ROWS_PER_WAVE * M_DIM);
        // Generic pointer to LDS: low 32 bits are the LDS byte offset.
        const uint32_t lds_base = (uint32_t)(uintptr_t)(&tile[0]);
        const uint32_t lds_addr = lds_base + (uint32_t)wave * ROWS_PER_WAVE * M_DIM * 4u;

        // ---- D# group 0 (128b): count | lds_addr | global_addr | type=2 ----
        u32x4 g0;
        g0.x = 1u;                                              // count=1, user mode
        g0.y = lds_addr;                                        // bits 63:32
        g0.z = (uint32_t)gaddr;                                 // global_addr[31:0]
        g0.w = (uint32_t)((gaddr >> 32) & 0x01FFFFFFu)          // global_addr[56:32]
             | (2u << 30);                                      // type = 2 (image)

        // ---- D# group 1 (256b): sizes/strides for a 2D 128x16 tile ---------
        u32x8 g1;
        g1.s0 = (2u << 16);                 // workgroup_mask=0, data_size=2 -> 4B
        g1.s1 = ((uint32_t)M_DIM) << 16;    // tensor_dim0[15:0]=128 (bits 63:48)
        g1.s2 = ((uint32_t)ROWS_PER_WAVE) << 16; // tensor_dim0 hi=0 | tensor_dim1 lo=16
        g1.s3 = ((uint32_t)M_DIM) << 16;    // tensor_dim1 hi=0 | tile_dim0=128
        g1.s4 = (uint32_t)ROWS_PER_WAVE;    // tile_dim1=16, tile_dim2=0
        g1.s5 = (uint32_t)M_DIM;            // tensor_dim0_stride[31:0]=128 elems
        g1.s6 = 0u;                         // stride0 hi | tensor_dim1_stride lo
        g1.s7 = 0u;

        // 2-group form: VADDR2/VADDR3 = NULL (2D tensor).
        asm volatile("tensor_load_to_lds %0, %1"
                     :: "s"(g0), "s"(g1) : "memory");
        __builtin_amdgcn_s_wait_tensorcnt(0);
    }
    __syncthreads();   // all 8 slices resident in LDS

    // ---------------- Per-(b,c) uniform shift decomposition -----------------
    const float* pb = pos + (size_t)b * 8;   // [ox0..ox3, oy0..oy3]
    int   X0[C_DIM], Y0[C_DIM];
    float wx[C_DIM], wy[C_DIM];
#pragma unroll
    for (int c = 0; c < C_DIM; ++c) {
        const float ox = pb[c];
        const float oy = pb[4 + c];
        const float fx = floorf(ox);
        const float fy = floorf(oy);
        wx[c] = ox - fx;
        wy[c] = oy - fy;
        // |shift| < 20 keeps these in [12,51]; clamp is free insurance so the
        // 65x65 window [X0, X0+64] stays inside the 128-wide tile.
        X0[c] = min(max(C0 + (int)fx, 0), M_DIM - N_DIM - 1);
        Y0[c] = min(max(C0 + (int)fy, 0), M_DIM - N_DIM - 1);
    }

    // ---------------- 4-tap stencil from LDS, float4 coalesced stores -------
    float4* __restrict__ out4 = (float4*)(out + (size_t)b * (N_DIM * N_DIM * C_DIM));
#pragma unroll
    for (int it = 0; it < (N_DIM * N_DIM) / 256; ++it) {   // 16 sites per thread
        const int p = it * 256 + tid;
        const int y = p >> 6;
        const int x = p & (N_DIM - 1);
        float v[C_DIM];
#pragma unroll
        for (int c = 0; c < C_DIM; ++c) {
            const float* row0 = &tile[(Y0[c] + y) * M_DIM + X0[c] + x];
            const float* row1 = row0 + M_DIM;
            const float v00 = row0[0], v01 = row0[1];
            const float v10 = row1[0], v11 = row1[1];
            const float h0 = fmaf(wx[c], v01 - v00, v00);
            const float h1 = fmaf(wx[c], v11 - v10, v10);
            v[c] = fmaf(wy[c], h1 - h0, h0);
        }
        float4 r; r.x = v[0]; r.y = v[1]; r.z = v[2]; r.w = v[3];
        out4[p] = r;
    }
}

extern "C" void kernel_launch(void* const* d_in, const int* in_sizes, int n_in,
                              void* d_out, int out_size, void* d_ws, size_t ws_size,
                              hipStream_t stream) {
    (void)n_in; (void)out_size; (void)d_ws; (void)ws_size;
    const float* img = (const float*)d_in[0];   // padded_obj (B,128,128,1) f32
    const float* pos = (const float*)d_in[1];   // positions  (B,1,2,4)    f32
    // d_in[2] is the scalar N == 64, baked into the kernel.
    float* out = (float*)d_out;                 // (B,64,64,4) f32

    const int B = in_sizes[0] / (M_DIM * M_DIM);
    extract_patches_tdm_kernel<<<B, 256, 0, stream>>>(img, pos, out);
}